// DiscriminativeLoss_66477503807594
// MI455X (gfx1250) — compile-verified
//
#include <hip/hip_runtime.h>
#include <hip/hip_bf16.h>
#include <stdint.h>

typedef __attribute__((ext_vector_type(2))) float v2f;
typedef __attribute__((ext_vector_type(8))) float v8f;

#define NB 8                 // batches
#define ND 16                // embedding dims
#define NPT (512 * 512)      // points per batch
#define NK 8                 // clusters
#define BLKS_PER_B 64
#define THREADS 256
#define WAVES 8              // 256 / wave32
#define CHUNKS 16            // 32-point chunks per wave (512 pts / wave)
#define NBUF 4               // async LDS double^2-buffering (prefetch depth 3)

// workspace layout (floats)
#define WS_SUMS 0                       // NB * NK * 16 (cluster sums)
#define WS_CNT  (NB * NK * 16)          // NB * 16     (cluster counts)
#define WS_CTR  (WS_CNT + NB * 16)      // NB * NK * 16 (centers)
#define WS_ACC  (WS_CTR + NB * NK * 16) // [0]=var [1]=dist [2]=reg
#define WS_TOT  (WS_ACC + 8)

// ------------------------------------------------ gfx1250 async mem -> LDS
__device__ __forceinline__ void async_ld_b128(uint32_t lds_addr, uint32_t voff,
                                              uint64_t sbase) {
  asm volatile("global_load_async_to_lds_b128 %0, %1, %2"
               :: "v"(lds_addr), "v"(voff), "s"(sbase) : "memory");
}
__device__ __forceinline__ void async_ld_b32(uint32_t lds_addr, uint32_t voff,
                                             uint64_t sbase) {
  asm volatile("global_load_async_to_lds_b32 %0, %1, %2"
               :: "v"(lds_addr), "v"(voff), "s"(sbase) : "memory");
}
__device__ __forceinline__ uint32_t lds_off(const void* p) {
  return (uint32_t)(size_t)p;  // generic shared ptr: low 32 bits = LDS addr
}

// ---------------------------------------------------------------- k0: zero ws
__global__ void k0_zero(float* __restrict__ ws) {
  int t = blockIdx.x * blockDim.x + threadIdx.x;
  for (int i = t; i < WS_TOT; i += blockDim.x * gridDim.x) ws[i] = 0.0f;
}

// ------------------------------------------- k1: cluster sums/counts via WMMA
// D[16 clusters x 16 dims] += onehot[16 x 4pts] * X[4pts x 16 dims]  (f32 WMMA)
// 4-deep async global->LDS pipeline, no block barriers in the loop.
__global__ __launch_bounds__(THREADS) void k1_sums(
    const float* __restrict__ data, const int* __restrict__ labels,
    float* __restrict__ ws) {
  __shared__ __align__(16) float tile[WAVES][NBUF][ND * 36];  // [dim][pt], pad 36
  __shared__ int   labs[WAVES][NBUF][32];
  __shared__ float sSum[NK][16];
  __shared__ float sCnt[NK];

  const int b    = blockIdx.x / BLKS_PER_B;
  const int blk  = blockIdx.x % BLKS_PER_B;
  const int tid  = threadIdx.x;
  const int wave = tid >> 5;
  const int lane = tid & 31;
  const int half = lane >> 4;   // which 16-lane half
  const int colm = lane & 15;   // dim col for B / cluster row for A

  if (tid < NK * 16) sSum[tid >> 4][tid & 15] = 0.0f;
  if (tid < NK) sCnt[tid] = 0.0f;
  __syncthreads();

  const int waveId = blk * WAVES + wave;
  const int base = waveId * (CHUNKS * 32);  // 512 points per wave

  const float* dbase = data + (size_t)b * ND * NPT;
  const int*   lbase = labels + (size_t)b * NPT;
  const uint64_t dptr = (uint64_t)(size_t)dbase;
  const uint64_t lptr = (uint64_t)(size_t)lbase;

  v8f acc = {0.f, 0.f, 0.f, 0.f, 0.f, 0.f, 0.f, 0.f};
  int cnt[NK];
#pragma unroll
  for (int k = 0; k < NK; ++k) cnt[k] = 0;

  // issue one chunk's 5 async transfers (4 x B128 tile + 1 x B32 label)
  auto issue = [&](int c) {
    const int buf = c & (NBUF - 1);
    const int n0 = base + c * 32;
#pragma unroll
    for (int r = 0; r < 4; ++r) {
      int idx = r * 32 + lane;
      int dim = idx >> 3;
      int p4  = (idx & 7) * 4;
      uint32_t voff = (uint32_t)(((uint32_t)dim * NPT + (uint32_t)(n0 + p4)) * 4u);
      async_ld_b128(lds_off(&tile[wave][buf][dim * 36 + p4]), voff, dptr);
    }
    async_ld_b32(lds_off(&labs[wave][buf][lane]),
                 (uint32_t)((uint32_t)(n0 + lane) * 4u), lptr);
  };

  // consume chunk c (tile is resident in LDS)
  auto compute = [&](int c) {
    const int buf = c & (NBUF - 1);
    const int myLab = labs[wave][buf][lane];
#pragma unroll
    for (int k = 0; k < NK; ++k) cnt[k] += (myLab == k) ? 1 : 0;
#pragma unroll
    for (int g = 0; g < 8; ++g) {
      const int p = g * 4 + 2 * half;  // K-rows {p, p+1} for this half
      v2f bv = *(const v2f*)&tile[wave][buf][colm * 36 + p];
      const int l0 = labs[wave][buf][p];
      const int l1 = labs[wave][buf][p + 1];
      v2f av;
      av.x = (l0 == colm) ? 1.0f : 0.0f;
      av.y = (l1 == colm) ? 1.0f : 0.0f;
      acc = __builtin_amdgcn_wmma_f32_16x16x4_f32(
          false, av, false, bv, (short)0, acc, false, false);
    }
  };

  // prologue: 3 chunks in flight
  issue(0); issue(1); issue(2);
  // steady state: issue c+3, wait until chunk c landed (<=15 outstanding)
  for (int c = 0; c < CHUNKS - 3; ++c) {
    issue(c + 3);
    asm volatile("s_wait_asynccnt 0xf" ::: "memory");
    compute(c);
  }
  // drain
  asm volatile("s_wait_asynccnt 0xa" ::: "memory");
  compute(CHUNKS - 3);
  asm volatile("s_wait_asynccnt 0x5" ::: "memory");
  compute(CHUNKS - 2);
  asm volatile("s_wait_asynccnt 0x0" ::: "memory");
  compute(CHUNKS - 1);

  __syncthreads();
  // C layout: lane l<16 holds rows 0..7 (col=l); lanes 16..31 rows 8..15 (zero)
  if (lane < 16) {
#pragma unroll
    for (int r = 0; r < 8; ++r) atomicAdd(&sSum[r][colm], acc[r]);
  }
#pragma unroll
  for (int k = 0; k < NK; ++k) atomicAdd(&sCnt[k], (float)cnt[k]);
  __syncthreads();

  if (tid < NK * 16) {
    int r = tid >> 4, j = tid & 15;
    atomicAdd(&ws[WS_SUMS + b * (NK * 16) + r * 16 + j], sSum[r][j]);
  }
  if (tid < NK) atomicAdd(&ws[WS_CNT + b * 16 + tid], sCnt[tid]);
}

// ------------------------------ k2: centers, pairwise-dist term, reg term
__global__ __launch_bounds__(256) void k2_centers(float* __restrict__ ws) {
  __shared__ float sC[NB][NK][17];
  const int tid = threadIdx.x;

  for (int idx = tid; idx < NB * NK * ND; idx += 256) {
    int b = idx >> 7, rem = idx & 127, k = rem >> 4, j = rem & 15;
    float cv = ws[WS_SUMS + b * (NK * 16) + k * 16 + j] /
               ws[WS_CNT + b * 16 + k];
    ws[WS_CTR + b * (NK * 16) + k * 16 + j] = cv;
    sC[b][k][j] = cv;
  }
  __syncthreads();

  for (int cidx = tid; cidx < NB * NK * NK; cidx += 256) {
    int b = cidx >> 6, i = (cidx >> 3) & 7, j = cidx & 7;
    if (i == j) continue;
    float s = 0.f;
#pragma unroll
    for (int t = 0; t < ND; ++t) {
      float dd = sC[b][i][t] - sC[b][j][t];
      s += dd * dd;
    }
    float cost = fmaxf(2.0f - sqrtf(s), 0.0f);  // DELTA_DIST = 2
    atomicAdd(&ws[WS_ACC + 1], cost * cost);
  }

  if (tid < NB * NK) {
    int b = tid >> 3, k = tid & 7;
    float s = 0.f;
#pragma unroll
    for (int t = 0; t < ND; ++t) s += sC[b][k][t] * sC[b][k][t];
    float r = fmaxf(sqrtf(s) - 4.0f, 0.0f);  // delta_reg = sqrt(16) = 4
    atomicAdd(&ws[WS_ACC + 2], r * r);
  }
}

// ------------------------------------------------- k3: variance term (pass 2)
__global__ __launch_bounds__(256) void k3_var(
    const float* __restrict__ data, const int* __restrict__ labels,
    float* __restrict__ ws) {
  __shared__ float sC[NK * 17];
  __shared__ float sInv[NK];
  __shared__ float red[256];

  const int b   = blockIdx.x / BLKS_PER_B;
  const int blk = blockIdx.x % BLKS_PER_B;
  const int tid = threadIdx.x;

  if (tid < NK * ND) {
    int k = tid >> 4, j = tid & 15;
    sC[k * 17 + j] = ws[WS_CTR + b * (NK * 16) + k * 16 + j];
  }
  if (tid < NK)
    sInv[tid] = 1.0f / ((float)NK * ws[WS_CNT + b * 16 + tid]);
  __syncthreads();

  const float* dbase = data + (size_t)b * ND * NPT;
  const int*   lbase = labels + (size_t)b * NPT;

  float acc = 0.0f;
  const int stride = BLKS_PER_B * 256;  // 16384
  for (int n = blk * 256 + tid; n < NPT; n += stride) {
    int lab = lbase[n];
    float s = 0.f;
#pragma unroll
    for (int j = 0; j < ND; ++j) {
      float x  = dbase[(size_t)j * NPT + n];
      float dd = x - sC[lab * 17 + j];
      s += dd * dd;
    }
    float e = fmaxf(sqrtf(s) - 1.0f, 0.0f);  // DELTA_VAR = 1
    acc += e * e * sInv[lab];
  }

  red[tid] = acc;
  __syncthreads();
  for (int off = 128; off > 0; off >>= 1) {
    if (tid < off) red[tid] += red[tid + off];
    __syncthreads();
  }
  if (tid == 0) atomicAdd(&ws[WS_ACC + 0], red[0]);
}

// ------------------------------------------------------------- k4: finalize
__global__ void k4_final(const float* __restrict__ ws, float* __restrict__ out) {
  out[0] = ws[WS_ACC + 0] / (float)NB +
           ws[WS_ACC + 1] / ((float)NB * NK * (NK - 1)) +
           ws[WS_ACC + 2] / ((float)NB * NK);
}

extern "C" void kernel_launch(void* const* d_in, const int* in_sizes, int n_in,
                              void* d_out, int out_size, void* d_ws, size_t ws_size,
                              hipStream_t stream) {
  const float* data   = (const float*)d_in[0];
  const int*   labels = (const int*)d_in[1];
  float* ws  = (float*)d_ws;
  float* out = (float*)d_out;
  (void)in_sizes; (void)n_in; (void)out_size; (void)ws_size;

  k0_zero<<<4, 256, 0, stream>>>(ws);
  k1_sums<<<NB * BLKS_PER_B, THREADS, 0, stream>>>(data, labels, ws);
  k2_centers<<<1, 256, 0, stream>>>(ws);
  k3_var<<<NB * BLKS_PER_B, 256, 0, stream>>>(data, labels, ws);
  k4_final<<<1, 1, 0, stream>>>(ws, out);
}